// MLP_64570538328612
// MI455X (gfx1250) — compile-verified
//
#include <hip/hip_runtime.h>
#include <cstdint>
#include <cstddef>

typedef __attribute__((ext_vector_type(2))) float v2f;
typedef __attribute__((ext_vector_type(8))) float v8f;

#define GXD 432
#define GYD 496
#define GZD 32
#define BD  2
#define PCH 16      // point feature channels
#define FCH 64      // sparse feature channels
#define ICH 80      // PCH + FCH
#define H1  40      // ic/2
#define NB  32      // z bins
#define OCD 64      // output channels
#define H4  256     // 4*OC
#define NSLOT (BD*GXD*GYD)   // 428544 pillar slots
#define EPSB 1e-3f

// ---------------- occupancy mark / rank / inv ----------------

__global__ void k_mark(const int* __restrict__ pmc, int* __restrict__ occ, int n) {
  int i = blockIdx.x * blockDim.x + threadIdx.x;
  if (i < n) {
    int code = pmc[i];
    int hi = code / (GYD * GZD);     // b*GX + px
    int py = code % (GYD * GZD);     // py (< GY)
    occ[hi * GYD + py] = 1;
  }
}

// single-workgroup exclusive scan over NSLOT elements (rank of occupied slots)
__global__ __launch_bounds__(1024) void k_scan(const int* __restrict__ occ,
                                               int* __restrict__ rank, int S) {
  __shared__ int buf[1024];
  int tid = threadIdx.x;
  int carry = 0;
  for (int base = 0; base < S; base += 1024) {
    int i = base + tid;
    int v = (i < S) ? occ[i] : 0;
    buf[tid] = v;
    __syncthreads();
    for (int off = 1; off < 1024; off <<= 1) {
      int t = (tid >= off) ? buf[tid - off] : 0;
      __syncthreads();
      buf[tid] += t;
      __syncthreads();
    }
    if (i < S) rank[i] = carry + buf[tid] - v;   // exclusive
    carry += buf[1023];
    __syncthreads();
  }
}

__global__ void k_inv(const int* __restrict__ pmc, const int* __restrict__ rank,
                      int* __restrict__ inv, int n) {
  int i = blockIdx.x * blockDim.x + threadIdx.x;
  if (i < n) {
    int code = pmc[i];
    int hi = code / (GYD * GZD);
    int py = code % (GYD * GZD);
    inv[i] = rank[hi * GYD + py];
  }
}

// ---------------- WMMA GEMMs (fp32, 16x16x4) ----------------
// Block = 8 waves; all waves share one 16-column tile (tn), stride row tiles.
// B column-tile staged in LDS as ldsB[k*16 + n] (bank-conflict-free, zero-filled
// for OOB columns so the K-loop is branchless). Lane l: m/n = l&15,
// k-group kg = 2*(l>>4). D: vgpr r -> row r + 8*(l>>4), col l&15.

// Fused stage-1 GEMM: feat = [pts[:,1:17] | sparse[inv]]  (K=80)  @ w1[80x40]
__global__ __launch_bounds__(256)
void k_gemm1(const float* __restrict__ pts, const float* __restrict__ sparse,
             const int* __restrict__ inv, const float* __restrict__ W,
             float* __restrict__ out, float* __restrict__ csum,
             float* __restrict__ csq, int tilesN, int tilesM) {
  const int P = H1;
  __shared__ float ldsB[ICH * 16];
  int wid = threadIdx.y, lane = threadIdx.x;
  int tid = wid * 32 + lane;
  int tn = blockIdx.x % tilesN;
  int tg = blockIdx.x / tilesN;
  for (int idx = tid; idx < ICH * 16; idx += 256) {
    int k = idx >> 4, n = idx & 15;
    int col = tn * 16 + n;
    ldsB[idx] = (col < P) ? W[(size_t)k * P + col] : 0.f;
  }
  __syncthreads();
  int tm = tg * 8 + wid;
  if (tm >= tilesM) return;
  int m16 = lane & 15;
  int kg = (lane >> 4) << 1;          // 0 or 2
  int row = tm * 16 + m16;
  const float* prow = pts + (size_t)row * 17 + 1 + kg;
  const float* srow = sparse + (size_t)inv[row] * FCH - PCH + kg;  // index by k>=16
  const float* brow = ldsB + kg * 16 + m16;
  v8f d = {};
#pragma unroll
  for (int k0 = 0; k0 < PCH; k0 += 4) {          // point-channel half
    v2f a; a.x = prow[k0]; a.y = prow[k0 + 1];
    v2f b; b.x = brow[k0 * 16]; b.y = brow[(k0 + 1) * 16];
    d = __builtin_amdgcn_wmma_f32_16x16x4_f32(false, a, false, b, (short)0, d, false, false);
  }
#pragma unroll 4
  for (int k0 = PCH; k0 < ICH; k0 += 4) {        // sparse-feature half (8B aligned)
    v2f a = *(const v2f*)(srow + k0);
    v2f b; b.x = brow[k0 * 16]; b.y = brow[(k0 + 1) * 16];
    d = __builtin_amdgcn_wmma_f32_16x16x4_f32(false, a, false, b, (short)0, d, false, false);
  }
  int col = tn * 16 + m16;
  int rb = tm * 16 + ((lane >> 4) << 3);
  if (col < P) {
    float s = 0.f, s2 = 0.f;
#pragma unroll
    for (int r = 0; r < 8; r++) {
      float v = d[r];
      out[(size_t)(rb + r) * P + col] = v;
      s += v; s2 += v * v;
    }
    atomicAdd(&csum[col], s);
    atomicAdd(&csq[col], s2);
  }
}

// Generic GEMM C[n x P] = A[n x KD] @ B[KD x P], per-column (optionally
// mask-weighted) sum / sum-of-squares accumulation for BN statistics.
template <int KD>
__global__ __launch_bounds__(256)
void k_gemm(const float* __restrict__ A, const float* __restrict__ B,
            float* __restrict__ C, float* __restrict__ csum,
            float* __restrict__ csq, const float* __restrict__ wmask,
            int P, int tilesN, int tilesM) {
  __shared__ float ldsB[KD * 16];
  int wid = threadIdx.y, lane = threadIdx.x;
  int tid = wid * 32 + lane;
  int tn = blockIdx.x % tilesN;
  int tg = blockIdx.x / tilesN;
  for (int idx = tid; idx < KD * 16; idx += 256) {
    int k = idx >> 4, n = idx & 15;
    int col = tn * 16 + n;
    ldsB[idx] = (col < P) ? B[(size_t)k * P + col] : 0.f;
  }
  __syncthreads();
  int tm = tg * 8 + wid;
  if (tm >= tilesM) return;
  int m16 = lane & 15;
  int kg = (lane >> 4) << 1;
  int row = tm * 16 + m16;
  const float* arow = A + (size_t)row * KD + kg;  // 8-byte aligned
  const float* brow = ldsB + kg * 16 + m16;
  v8f d = {};
#pragma unroll 8
  for (int k0 = 0; k0 < KD; k0 += 4) {
    v2f a = *(const v2f*)(arow + k0);
    v2f b; b.x = brow[k0 * 16]; b.y = brow[(k0 + 1) * 16];
    d = __builtin_amdgcn_wmma_f32_16x16x4_f32(false, a, false, b, (short)0, d, false, false);
  }
  int col = tn * 16 + m16;
  int rb = tm * 16 + ((lane >> 4) << 3);
  if (col < P) {
    float s = 0.f, s2 = 0.f;
#pragma unroll
    for (int r = 0; r < 8; r++) {
      float v = d[r];
      C[(size_t)(rb + r) * P + col] = v;
      float w = wmask ? wmask[rb + r] : 1.f;
      s += v * w; s2 += v * v * w;
    }
    atomicAdd(&csum[col], s);
    atomicAdd(&csq[col], s2);
  }
}

// ---------------- BN helpers ----------------

__global__ void k_finalize(const float* __restrict__ csum, const float* __restrict__ csq,
                           float* __restrict__ mean, float* __restrict__ istd,
                           int P, const float* __restrict__ cntPtr, float cntConst) {
  int c = threadIdx.x;
  if (c < P) {
    float cnt = cntPtr ? *cntPtr : cntConst;
    float m = csum[c] / cnt;
    float v = csq[c] / cnt - m * m;   // biased variance
    mean[c] = m;
    istd[c] = rsqrtf(v + EPSB);
  }
}

__global__ void k_bnrelu(float* __restrict__ X, const float* __restrict__ mean,
                         const float* __restrict__ istd, const float* __restrict__ g,
                         const float* __restrict__ b, int P, size_t total) {
  size_t i = (size_t)blockIdx.x * blockDim.x + threadIdx.x;
  if (i < total) {
    int c = (int)(i % (size_t)P);
    float v = (X[i] - mean[c]) * istd[c] * g[c] + b[c];
    X[i] = v > 0.f ? v : 0.f;
  }
}

// ---------------- voxel scatter (BN2 fused) ----------------

__global__ void k_scatter(const float* __restrict__ pts, const float* __restrict__ ppre,
                          const float* __restrict__ mean, const float* __restrict__ istd,
                          const float* __restrict__ g, const float* __restrict__ bb,
                          const int* __restrict__ rank, const float* __restrict__ pcr,
                          const float* __restrict__ vsz, float* __restrict__ ssum,
                          float* __restrict__ scnt, int n) {
  int gid = blockIdx.x * blockDim.x + threadIdx.x;
  if (gid >= n * PCH) return;
  int i = gid >> 4;
  int c = gid & 15;
  float v = ppre[(size_t)i * PCH + c];
  v = (v - mean[c]) * istd[c] * g[c] + bb[c];
  v = v > 0.f ? v : 0.f;
  const float* p = pts + (size_t)i * 17;
  int b  = (int)p[0];
  int vx = (int)floorf((p[1] - pcr[0]) / vsz[0]);
  int vy = (int)floorf((p[2] - pcr[1]) / vsz[1]);
  int vz = (int)floorf((p[3] - pcr[2]) / vsz[2]);
  int pr = rank[(b * GXD + vx) * GYD + vy];
  atomicAdd(&ssum[((size_t)pr * NB + vz) * PCH + c], v);
  if (c == 0) atomicAdd(&scnt[(size_t)pr * NB + vz], 1.f);
}

// ---------------- bin means + channel/spatial attention + occ mask ----------------
// 4 pillars per 128-thread block; lane == z-bin.

__global__ void k_attn(const float* __restrict__ ssum, const float* __restrict__ scnt,
                       const float* __restrict__ caw, const float* __restrict__ cab,
                       const float* __restrict__ saw, const float* __restrict__ sab,
                       float* __restrict__ flat, float* __restrict__ occw,
                       float* __restrict__ wsum, int M) {
  __shared__ float tile[4][NB * PCH];
  __shared__ float cav[4][PCH];
  __shared__ int   occ[4][NB];
  int ty = threadIdx.y, lane = threadIdx.x;
  int p = blockIdx.x * 4 + ty;
  if (p >= M) return;
  float cnt = scnt[(size_t)p * NB + lane];
  float inv = cnt > 0.f ? 1.f / cnt : 0.f;
#pragma unroll
  for (int c = 0; c < PCH; c++)
    tile[ty][lane * PCH + c] = ssum[((size_t)p * NB + lane) * PCH + c] * inv;
  occ[ty][lane] = cnt > 0.f ? 1 : 0;
  __syncthreads();
  if (lane == 0) {
    int s = 0;
    for (int b = 0; b < NB; b++) s += occ[ty][b];
    float w = (s >= 2) ? 1.f : 0.f;
    occw[p] = w;
    atomicAdd(wsum, w);
  }
  if (lane < PCH) {
    float s = 0.f;
    for (int b = 0; b < NB; b++) s += tile[ty][b * PCH + lane] * caw[b];
    cav[ty][lane] = s + cab[0];
  }
  __syncthreads();
  float sa = 0.f;
  float vals[PCH];
#pragma unroll
  for (int c = 0; c < PCH; c++) {
    float v = tile[ty][lane * PCH + c] * cav[ty][c];
    vals[c] = v;
    sa += v * saw[c];
  }
  sa += sab[0];
#pragma unroll
  for (int c = 0; c < PCH; c++)
    flat[((size_t)p * NB + lane) * PCH + c] = vals[c] * sa;
}

// ---------------- final residual ----------------

__global__ void k_final(const float* __restrict__ opre, const float* __restrict__ mean,
                        const float* __restrict__ istd, const float* __restrict__ g,
                        const float* __restrict__ bb, const float* __restrict__ occw,
                        const float* __restrict__ sparse, float* __restrict__ out, int M) {
  int gid = blockIdx.x * blockDim.x + threadIdx.x;
  if (gid >= M * OCD) return;
  int p = gid / OCD, c = gid - p * OCD;
  float v = (opre[gid] - mean[c]) * istd[c] * g[c] + bb[c];
  v = v > 0.f ? v : 0.f;
  out[gid] = sparse[gid] + occw[p] * v;
}

// ---------------- launch ----------------

extern "C" void kernel_launch(void* const* d_in, const int* in_sizes, int n_in,
                              void* d_out, int out_size, void* d_ws, size_t ws_size,
                              hipStream_t stream) {
  const float* pts    = (const float*)d_in[0];
  const int*   pmc    = (const int*)d_in[1];
  const float* sparse = (const float*)d_in[2];
  const float* pcr    = (const float*)d_in[3];
  const float* vsz    = (const float*)d_in[4];
  const float* w1     = (const float*)d_in[5];
  const float* g1     = (const float*)d_in[6];
  const float* b1     = (const float*)d_in[7];
  const float* w2     = (const float*)d_in[8];
  const float* g2     = (const float*)d_in[9];
  const float* b2     = (const float*)d_in[10];
  const float* caw    = (const float*)d_in[11];
  const float* cab    = (const float*)d_in[12];
  const float* saw    = (const float*)d_in[13];
  const float* sab    = (const float*)d_in[14];
  const float* bw1    = (const float*)d_in[15];
  const float* bg1    = (const float*)d_in[16];
  const float* bb1    = (const float*)d_in[17];
  const float* bw2    = (const float*)d_in[18];
  const float* bg2    = (const float*)d_in[19];
  const float* bb2    = (const float*)d_in[20];

  const int N = in_sizes[0] / 17;        // 400000
  const int M = in_sizes[2] / FCH;       // 60000
  float* out = (float*)d_out;

  size_t off = 0;
  auto alloc = [&](size_t bytes) -> void* {
    void* p = (char*)d_ws + off;
    off = (off + bytes + 255) & ~(size_t)255;
    return p;
  };
  int*   inv   = (int*)alloc((size_t)N * 4);
  int*   occ   = (int*)alloc((size_t)NSLOT * 4);
  int*   rank  = (int*)alloc((size_t)NSLOT * 4);
  float* hpre  = (float*)alloc((size_t)N * H1 * 4);
  float* ppre  = (float*)alloc((size_t)N * PCH * 4);
  float* src   = (float*)alloc((size_t)M * NB * PCH * 4);  // sums -> means -> flat
  float* scnt  = (float*)alloc((size_t)M * NB * 4);
  float* occw  = (float*)alloc((size_t)M * 4);
  float* h2    = (float*)alloc((size_t)M * H4 * 4);
  float* opre  = (float*)alloc((size_t)M * OCD * 4);
  float* stats = (float*)alloc(1024 * 4);   // accumulators
  float* ms    = (float*)alloc(1024 * 4);   // mean / inv-std
  // stats layout
  float* sum1 = stats + 0,   *sq1 = stats + 64;
  float* sum2 = stats + 128, *sq2 = stats + 192;
  float* sum3 = stats + 256, *sq3 = stats + 512;
  float* sum4 = stats + 768, *sq4 = stats + 832;
  float* wsum = stats + 896;
  float* m1 = ms + 0,   *is1 = ms + 64;
  float* m2 = ms + 128, *is2 = ms + 192;
  float* m3 = ms + 256, *is3 = ms + 512;
  float* m4 = ms + 768, *is4 = ms + 832;

  // zero accumulators
  hipMemsetAsync(occ,   0, (size_t)NSLOT * 4, stream);
  hipMemsetAsync(src,   0, (size_t)M * NB * PCH * 4, stream);
  hipMemsetAsync(scnt,  0, (size_t)M * NB * 4, stream);
  hipMemsetAsync(stats, 0, 1024 * 4, stream);

  dim3 gblk(32, 8);   // 8 waves -> 8 row tiles sharing one staged B column tile

  // pillar rank table
  k_mark<<<(N + 255) / 256, 256, 0, stream>>>(pmc, occ, N);
  k_scan<<<1, 1024, 0, stream>>>(occ, rank, NSLOT);
  k_inv<<<(N + 255) / 256, 256, 0, stream>>>(pmc, rank, inv, N);

  // stage 1: feat @ w1  (K=80, P=40)
  {
    int tilesM = N / 16, tilesN = 3;
    int grid = tilesN * ((tilesM + 7) / 8);
    k_gemm1<<<grid, gblk, 0, stream>>>(pts, sparse, inv, w1, hpre, sum1, sq1, tilesN, tilesM);
  }
  k_finalize<<<1, 64, 0, stream>>>(sum1, sq1, m1, is1, H1, nullptr, (float)N);
  k_bnrelu<<<(int)(((size_t)N * H1 + 255) / 256), 256, 0, stream>>>(hpre, m1, is1, g1, b1, H1, (size_t)N * H1);

  // stage 1b: h @ w2  (K=40, P=16)
  {
    int tilesM = N / 16, tilesN = 1;
    int grid = tilesN * ((tilesM + 7) / 8);
    k_gemm<H1><<<grid, gblk, 0, stream>>>(hpre, w2, ppre, sum2, sq2, nullptr, PCH, tilesN, tilesM);
  }
  k_finalize<<<1, 64, 0, stream>>>(sum2, sq2, m2, is2, PCH, nullptr, (float)N);

  // voxel scatter-mean into (pillar, z) bins; BN2+relu fused
  k_scatter<<<(N * PCH + 255) / 256, 256, 0, stream>>>(pts, ppre, m2, is2, g2, b2, rank, pcr, vsz, src, scnt, N);

  // bin means + attention + occupancy mask
  k_attn<<<(M + 3) / 4, dim3(32, 4), 0, stream>>>(src, scnt, caw, cab, saw, sab, src, occw, wsum, M);

  // stage 5a: flat @ bw1  (K=512, P=256) with masked BN stats
  {
    int tilesM = M / 16, tilesN = H4 / 16;
    int grid = tilesN * ((tilesM + 7) / 8);
    k_gemm<NB * PCH><<<grid, gblk, 0, stream>>>(src, bw1, h2, sum3, sq3, occw, H4, tilesN, tilesM);
  }
  k_finalize<<<1, 256, 0, stream>>>(sum3, sq3, m3, is3, H4, wsum, 0.f);
  k_bnrelu<<<(int)(((size_t)M * H4 + 255) / 256), 256, 0, stream>>>(h2, m3, is3, bg1, bb1, H4, (size_t)M * H4);

  // stage 5b: h2 @ bw2  (K=256, P=64) with masked BN stats
  {
    int tilesM = M / 16, tilesN = OCD / 16;
    int grid = tilesN * ((tilesM + 7) / 8);
    k_gemm<H4><<<grid, gblk, 0, stream>>>(h2, bw2, opre, sum4, sq4, occw, OCD, tilesN, tilesM);
  }
  k_finalize<<<1, 64, 0, stream>>>(sum4, sq4, m4, is4, OCD, wsum, 0.f);

  // residual add
  k_final<<<(M * OCD + 255) / 256, 256, 0, stream>>>(opre, m4, is4, bg2, bb2, occw, sparse, out, M);
}